// CustomGlobalAttention_61761629716510
// MI455X (gfx1250) — compile-verified
//
#include <hip/hip_runtime.h>
#include <math.h>

#define NUM_GRAPHS 64
#define D 512
#define EPSF 1e-16f

typedef float v2f __attribute__((ext_vector_type(2)));
typedef float v8f __attribute__((ext_vector_type(8)));

// ---- order-preserving float<->uint map (for atomicMax on floats) ----
__device__ __forceinline__ unsigned ford(float f) {
    unsigned u = __float_as_uint(f);
    return (u & 0x80000000u) ? ~u : (u | 0x80000000u);
}
__device__ __forceinline__ float funord(unsigned o) {
    return (o & 0x80000000u) ? __uint_as_float(o & 0x7FFFFFFFu)
                             : __uint_as_float(~o);
}

// ---- K0: zero scratch accumulators (ws is poisoned 0xAA; must reset per call) ----
__global__ void k0_zero(unsigned* m_ord, float* s, float* y) {
    int i = blockIdx.x * 256 + threadIdx.x;
    if (i < NUM_GRAPHS) { m_ord[i] = 0u; s[i] = 0.0f; }
    if (i < NUM_GRAPHS * D) y[i] = 0.0f;
}

// ---- K1: gate = nan_to_num(x @ gate_w + gate_b); segment max via atomicMax ----
// one wave32 per row, 8 rows per 256-thread block
__global__ void k1_gate(const float* __restrict__ x, const int* __restrict__ batch,
                        const float* __restrict__ gw, const float* __restrict__ gb,
                        float* __restrict__ gate, unsigned* __restrict__ m_ord, int N) {
    __shared__ int      sh_b[8];
    __shared__ unsigned sh_o[8];
    int wave = threadIdx.x >> 5;
    int lane = threadIdx.x & 31;
    int row  = blockIdx.x * 8 + wave;
    if (row < N) {
        const float4* xr  = (const float4*)(x + (long)row * D);
        const float4* gwv = (const float4*)gw;
        float acc = 0.0f;
#pragma unroll
        for (int j = 0; j < 4; ++j) {          // 32 lanes * 4 * float4 = 512
            float4 v = xr[lane + 32 * j];
            float4 w = gwv[lane + 32 * j];
            acc += v.x * w.x + v.y * w.y + v.z * w.z + v.w * w.w;
        }
#pragma unroll
        for (int off = 16; off; off >>= 1) acc += __shfl_xor(acc, off, 32);
        if (lane == 0) {
            float g = acc + gb[0];
            if (isnan(g))      g = 0.0f;
            else if (isinf(g)) g = (g > 0.0f) ? 1e6f : -1e6f;
            gate[row]  = g;
            sh_b[wave] = batch[row];
            sh_o[wave] = ford(g);
        }
    } else if (lane == 0) {
        sh_b[wave] = -1;
    }
    __syncthreads();
    if (threadIdx.x == 0) {                    // merge <=8 (sorted) entries -> few atomics
        int cb = sh_b[0]; unsigned co = sh_o[0];
        for (int w = 1; w < 8; ++w) {
            int b = sh_b[w];
            if (b < 0) break;
            if (b == cb) { unsigned o = sh_o[w]; co = (o > co) ? o : co; }
            else { atomicMax(&m_ord[cb], co); cb = b; co = sh_o[w]; }
        }
        if (cb >= 0) atomicMax(&m_ord[cb], co);
    }
}

// ---- K2: s[g] = segment_sum(exp(gate - m[g])) with LDS pre-reduction ----
__global__ void k2_expsum(const float* __restrict__ gate, const int* __restrict__ batch,
                          const unsigned* __restrict__ m_ord, float* __restrict__ s, int N) {
    __shared__ float sh_s[NUM_GRAPHS];
    int t = threadIdx.x;
    if (t < NUM_GRAPHS) sh_s[t] = 0.0f;
    __syncthreads();
    long row = (long)blockIdx.x * 256 + t;
    if (row < N) {
        int b = batch[row];
        float e = expf(gate[row] - funord(m_ord[b]));
        atomicAdd(&sh_s[b], e);
    }
    __syncthreads();
    if (t < NUM_GRAPHS && sh_s[t] != 0.0f) atomicAdd(&s[t], sh_s[t]);
}

// ---- K3: attn_i = e_i/(s+eps) (written to d_out) ; y[g,:] += attn_i * x[i,:] ----
// 256-thread block handles 256 contiguous rows; thread t owns columns {2t,2t+1}
// register accumulator; flush to global y on (rare) segment change.
#define K3_ROWS 256
__global__ void k3_pool(const float* __restrict__ x, const int* __restrict__ batch,
                        const float* __restrict__ gate, const unsigned* __restrict__ m_ord,
                        const float* __restrict__ s, float* __restrict__ y,
                        float* __restrict__ attn_out, int N) {
    __shared__ float sh_attn[K3_ROWS];
    __shared__ int   sh_b[K3_ROWS];
    int t = threadIdx.x;
    long base = (long)blockIdx.x * K3_ROWS;
    long row = base + t;
    float a = 0.0f;
    int b = NUM_GRAPHS - 1;
    if (row < N) {
        b = batch[row];
        float e = expf(gate[row] - funord(m_ord[b]));
        a = e / (s[b] + EPSF);
        attn_out[row] = a;
    }
    sh_attn[t] = a;
    sh_b[t]    = b;
    __syncthreads();

    const int c0 = t * 2;
    float2 acc = make_float2(0.0f, 0.0f);
    int cur = sh_b[0];
    for (int r = 0; r < K3_ROWS; ++r) {
        int rb = sh_b[r];                      // uniform across block -> no divergence
        if (rb != cur) {
            atomicAdd(&y[cur * D + c0],     acc.x);
            atomicAdd(&y[cur * D + c0 + 1], acc.y);
            acc = make_float2(0.0f, 0.0f);
            cur = rb;
        }
        float av = sh_attn[r];
        if (av != 0.0f) {                      // uniform; also skips OOB tail rows
            float2 v = *(const float2*)(x + (base + r) * D + c0);
            acc.x += av * v.x;
            acc.y += av * v.y;
        }
    }
    atomicAdd(&y[cur * D + c0],     acc.x);
    atomicAdd(&y[cur * D + c0 + 1], acc.y);
}

// ---- K4: pooled = y @ nn_w^T + sa*nn_b via V_WMMA_F32_16X16X4_F32 ----
// M=64, N=512, K=512. 4x32 = 128 output 16x16 tiles, one wave32 per tile.
// A 16x4 f32 frag: lane l -> m=l&15, k-pair = 2*(l>>4); C/D: c[j] -> m=8*(l>>4)+j, n=l&15.
__global__ void k4_wmma(const float* __restrict__ y, const float* __restrict__ nn_w,
                        const float* __restrict__ nn_b, const float* __restrict__ s,
                        float* __restrict__ pooled) {
    int gid  = blockIdx.x * blockDim.x + threadIdx.x;
    int wave = gid >> 5;          // 0..127
    int lane = threadIdx.x & 31;
    int mt = wave >> 5;           // 0..3   (M tile)
    int nt = wave & 31;           // 0..31  (N tile)
    int half = lane >> 4;         // 0/1
    int l15  = lane & 15;
    int am = mt * 16 + l15;       // A row (m) for this lane
    int bn = nt * 16 + l15;       // B col (n) for this lane
    v8f c = {};
    for (int k = 0; k < D; k += 4) {
        int kb = k + half * 2;    // even -> 8B-aligned v2f loads
        v2f a = *(const v2f*)(y    + am * D + kb);   // A[m, kb..kb+1]
        v2f b = *(const v2f*)(nn_w + bn * D + kb);   // B[kb..kb+1, n] = nn_w[n, kb..]
        c = __builtin_amdgcn_wmma_f32_16x16x4_f32(
                /*neg_a=*/false, a, /*neg_b=*/false, b,
                /*c_mod=*/(short)0, c, /*reuse_a=*/false, /*reuse_b=*/false);
    }
    float bias_n = nn_b[bn];
#pragma unroll
    for (int j = 0; j < 8; ++j) {
        int m = mt * 16 + half * 8 + j;
        float sg = s[m];
        float sa = sg / (sg + EPSF);             // exact segment attn sum
        pooled[m * D + bn] = c[j] + sa * bias_n;
    }
}

extern "C" void kernel_launch(void* const* d_in, const int* in_sizes, int n_in,
                              void* d_out, int out_size, void* d_ws, size_t ws_size,
                              hipStream_t stream) {
    const float* x      = (const float*)d_in[0];
    const int*   batch  = (const int*)d_in[1];
    const float* gate_w = (const float*)d_in[2];
    const float* gate_b = (const float*)d_in[3];
    const float* nn_w   = (const float*)d_in[4];
    const float* nn_b   = (const float*)d_in[5];
    int N = in_sizes[1];

    // workspace layout (bytes)
    char* ws = (char*)d_ws;
    float*    gate  = (float*)ws;                                  // N floats
    size_t off = (size_t)N * sizeof(float);
    off = (off + 255) & ~(size_t)255;
    unsigned* m_ord = (unsigned*)(ws + off); off += 256;           // 64 u32
    float*    s     = (float*)(ws + off);    off += 256;           // 64 f32
    float*    y     = (float*)(ws + off);                          // 64*512 f32

    float* pooled   = (float*)d_out;                               // [64,512]
    float* attn_out = (float*)d_out + NUM_GRAPHS * D;              // [N]

    k0_zero<<<(NUM_GRAPHS * D + 255) / 256, 256, 0, stream>>>(m_ord, s, y);
    k1_gate<<<(N + 7) / 8, 256, 0, stream>>>(x, batch, gate_w, gate_b, gate, m_ord, N);
    k2_expsum<<<(N + 255) / 256, 256, 0, stream>>>(gate, batch, m_ord, s, N);
    k3_pool<<<(N + K3_ROWS - 1) / K3_ROWS, K3_ROWS, 0, stream>>>(
        x, batch, gate, m_ord, s, y, attn_out, N);
    k4_wmma<<<16, 256, 0, stream>>>(y, nn_w, nn_b, s, pooled);
}